// MTAD_GAT_23227183136799
// MI455X (gfx1250) — compile-verified
//
#include <hip/hip_runtime.h>

// MTAD-GAT forward for MI455X (gfx1250, wave32).
// All GEMMs use V_WMMA_F32_16X16X4_F32 (exact f32 semantics vs reference).
// Problem is tiny (~1.5 GFLOP, ~11MB) -> L2-resident, latency bound; the
// 200-step GRU recurrence runs in ONE persistent workgroup with h in LDS and
// w_hh fragments pre-staged in registers (38 k-steps x 2 VGPR per wave).
// GEMM inner loop is guard-free (clamped indices, K%4==0 at all call sites),
// pointer-incremental, and software-pipelined (loads for k+4 in flight
// during the WMMA for k).

typedef float v2f __attribute__((ext_vector_type(2)));
typedef float v8f __attribute__((ext_vector_type(8)));

#define LRELU_ALPHA 0.2f

__device__ __forceinline__ float sigmoidf_(float x) { return 1.0f / (1.0f + __expf(-x)); }

// ---------------------------------------------------------------------------
// Conv1d (same padding, KS=7) + bias + ReLU.
// x: (8,200,100) [b,t,i] -> xc: (8,200,100) [b,t,o]; also writes hcat[...,0:100].
// Branch-free taps: boundary handling folded into loop bounds.
// ---------------------------------------------------------------------------
__global__ void conv_relu_kernel(const float* __restrict__ x, const float* __restrict__ cw,
                                 const float* __restrict__ cb, float* __restrict__ xc,
                                 float* __restrict__ hcat)
{
    int idx = blockIdx.x * blockDim.x + threadIdx.x;
    if (idx >= 8 * 200 * 100) return;
    int o = idx % 100;
    int t = (idx / 100) % 200;
    int b = idx / 20000;
    float s = cb[o];
    const float* xb = x + b * 20000;
    const float* w  = cw + o * 700;            // conv_w: (O=100, I=100, KS=7)
    int lo = (t >= 3) ? 0 : 3 - t;             // valid tap range: 0 <= t+ks-3 < 200
    int hi = (t <= 196) ? 7 : 203 - t;
    for (int i = 0; i < 100; ++i) {
        const float* wi = w + i * 7;
        const float* xi = xb + i + (t - 3) * 100;
        for (int ks = lo; ks < hi; ++ks) s += xi[ks * 100] * wi[ks];
    }
    s = s > 0.f ? s : 0.f;
    xc[b * 20000 + t * 100 + o]   = s;
    hcat[b * 60000 + t * 300 + o] = s;         // h_cat[:, :, 0:100] = xc
}

// ---------------------------------------------------------------------------
// Generic batched f32 GEMM: C[b] = op(A[b]) @ op(B[b]) (+ bias per column).
// One wave per 16x16 C tile, V_WMMA_F32_16X16X4_F32, fragments loaded
// per-lane straight from L2 (data set is tiny & L2-resident on MI455X).
// Fragment layout (ISA 7.12.2): lanes0-15 hold K={0,1}, lanes16-31 K={2,3};
// C VGPR v: lanes0-15 -> M=v, lanes16-31 -> M=v+8.
// REQUIRES: K % 4 == 0 (true at all call sites: 200, 100, 300).
// M/N tails: clamped load indices (no EXEC divergence); padded rows/cols of C
// compute duplicates that are never stored.
// ---------------------------------------------------------------------------
template <int TA, int TB>
__global__ void gemm_wmma_kernel(const float* __restrict__ A, const float* __restrict__ Bm,
                                 const float* __restrict__ bias, float* __restrict__ C,
                                 int M, int N, int K, int lda, int ldb, int ldc,
                                 long sA, long sB, long sC)
{
    int b = blockIdx.y;
    const float* Ab = A + (long)b * sA;
    const float* Bb = Bm + (long)b * sB;
    float* Cb = C + (long)b * sC;
    int tilesN = (N + 15) >> 4;
    int tilesM = (M + 15) >> 4;
    int tile = blockIdx.x * (blockDim.x >> 5) + (threadIdx.x >> 5);
    if (tile >= tilesM * tilesN) return;               // wave-uniform exit
    int tM = tile / tilesN, tN = tile % tilesN;
    int lane = threadIdx.x & 31;
    int half = lane >> 4, r = lane & 15;
    int m0 = tM * 16, n0 = tN * 16;
    int row = m0 + r, col = n0 + r;
    int rowc = row < M ? row : M - 1;
    int colc = col < N ? col : N - 1;

    const float* aPtr = TA ? Ab + (long)(2 * half) * lda + rowc : Ab + (long)rowc * lda + 2 * half;
    const float* bPtr = TB ? Bb + (long)colc * ldb + 2 * half   : Bb + (long)(2 * half) * ldb + colc;
    const int  aOff  = TA ? lda : 1;                   // compile-time 1 when TA==0
    const int  bOff  = TB ? 1 : ldb;                   // compile-time 1 when TB==1
    const long aStep = TA ? (long)4 * lda : 4;
    const long bStep = TB ? 4 : (long)4 * ldb;

    v8f acc = {};
    v2f a0, b0;
    a0.x = aPtr[0]; a0.y = aPtr[aOff];
    b0.x = bPtr[0]; b0.y = bPtr[bOff];
    for (int k0 = 4; k0 < K; k0 += 4) {
        aPtr += aStep; bPtr += bStep;
        v2f a1, b1;                                    // next fragments in flight
        a1.x = aPtr[0]; a1.y = aPtr[aOff];
        b1.x = bPtr[0]; b1.y = bPtr[bOff];
        acc = __builtin_amdgcn_wmma_f32_16x16x4_f32(false, a0, false, b0, (short)0, acc, false, false);
        a0 = a1; b0 = b1;
    }
    acc = __builtin_amdgcn_wmma_f32_16x16x4_f32(false, a0, false, b0, (short)0, acc, false, false);

    bool colOk = col < N;
    float bcol = (bias != nullptr && colOk) ? bias[col] : 0.f;
    #pragma unroll
    for (int v = 0; v < 8; ++v) {
        int rr = m0 + v + 8 * half;
        if (colOk && rr < M) Cb[(long)rr * ldc + col] = acc[v] + bcol;
    }
}

// ---------------------------------------------------------------------------
// Fused GATv2 attention row: one block per (b, i).
//   e_ij = a . leaky(si_i + sj_j)   (lin_b pre-folded into si)   + bias[i,j]
//   out_i = sigmoid(softmax_j(e) @ v)   written through stride view into hcat.
// ---------------------------------------------------------------------------
__global__ void gat_attn_kernel(const float* __restrict__ si, const float* __restrict__ sj,
                                const float* __restrict__ avec, const float* __restrict__ bias,
                                const float* __restrict__ vsrc, float* __restrict__ out,
                                int N, int E, int D,
                                long sSI, long sV, int vJs, int vDs,
                                long sOutB, int outIs, int outDs)
{
    __shared__ float eL[256];
    __shared__ float pL[256];
    int b = blockIdx.y, i = blockIdx.x;
    int tid = threadIdx.x, lane = tid & 31, wid = tid >> 5;
    int nwaves = blockDim.x >> 5;
    const float* sirow = si + b * sSI + (long)i * E;
    const float* sjb   = sj + b * sSI;

    for (int j = wid; j < N; j += nwaves) {
        const float* sjrow = sjb + (long)j * E;
        float p = 0.f;
        for (int e = lane; e < E; e += 32) {
            float g = sirow[e] + sjrow[e];
            g = (g >= 0.f) ? g : LRELU_ALPHA * g;
            p += avec[e] * g;
        }
        #pragma unroll
        for (int off = 16; off > 0; off >>= 1) p += __shfl_xor(p, off, 32);
        if (lane == 0) eL[j] = p + bias[i * N + j];
    }
    __syncthreads();
    float mx = -3.0e38f;
    for (int j = 0; j < N; ++j) mx = fmaxf(mx, eL[j]);
    for (int j = tid; j < N; j += blockDim.x) pL[j] = __expf(eL[j] - mx);
    __syncthreads();
    float ssum = 0.f;
    for (int j = 0; j < N; ++j) ssum += pL[j];
    float inv = 1.0f / ssum;
    const float* vb = vsrc + b * sV;
    for (int d = tid; d < D; d += blockDim.x) {
        float accd = 0.f;
        for (int j = 0; j < N; ++j) accd += pL[j] * vb[(long)j * vJs + (long)d * vDs];
        out[b * sOutB + (long)i * outIs + (long)d * outDs] = sigmoidf_(accd * inv);
    }
}

// ---------------------------------------------------------------------------
// GRU: 200 sequential steps in ONE workgroup (29 waves = 928 threads).
// h (8x150, padded 16x152) lives in LDS; gh = h @ w_hh^T via WMMA per step;
// each wave owns one 16-wide N tile and keeps its 38 w_hh B-fragments in
// registers for all 200 steps (zero weight traffic in the serial loop).
// ---------------------------------------------------------------------------
__global__ void gru_wmma_kernel(const float* __restrict__ Gi, const float* __restrict__ w_hh,
                                const float* __restrict__ b_hh, float* __restrict__ out)
{
    __shared__ float h_s[16][152];    // rows 8-15 and cols 150-151 stay zero
    __shared__ float gh_s[8][464];
    int tid = threadIdx.x, lane = tid & 31, wid = tid >> 5;
    int half = lane >> 4, r = lane & 15;

    for (int k = tid; k < 16 * 152; k += blockDim.x) ((float*)h_s)[k] = 0.f;

    int n0 = wid * 16;
    int col = n0 + r;
    v2f bf[38];
    #pragma unroll
    for (int kk = 0; kk < 38; ++kk) {                   // B[k][n] = w_hh[n*150 + k]
        int k = 4 * kk + 2 * half;
        bf[kk].x = (col < 450 && k     < 150) ? w_hh[(long)col * 150 + k]     : 0.f;
        bf[kk].y = (col < 450 && k + 1 < 150) ? w_hh[(long)col * 150 + k + 1] : 0.f;
    }

    for (int t = 0; t < 200; ++t) {
        __syncthreads();                                // h_s ready
        v8f acc = {};
        #pragma unroll
        for (int kk = 0; kk < 38; ++kk) {
            int k = 4 * kk + 2 * half;
            v2f av; av.x = h_s[r][k]; av.y = h_s[r][k + 1];
            acc = __builtin_amdgcn_wmma_f32_16x16x4_f32(false, av, false, bf[kk], (short)0, acc, false, false);
        }
        if (half == 0) {                                // rows 0..7 = batch
            #pragma unroll
            for (int v = 0; v < 8; ++v) gh_s[v][n0 + r] = acc[v];
        }
        __syncthreads();                                // gh ready
        if (tid < 1200) {
            int bb = tid / 150, u = tid % 150;
            const float* gi = Gi + (long)bb * 90000 + (long)t * 450;
            float hr = gh_s[bb][u]       + b_hh[u];
            float hz = gh_s[bb][150 + u] + b_hh[150 + u];
            float hn = gh_s[bb][300 + u] + b_hh[300 + u];
            float rg = sigmoidf_(gi[u]       + hr);
            float zg = sigmoidf_(gi[150 + u] + hz);
            float ng = tanhf(gi[300 + u] + rg * hn);
            h_s[bb][u] = (1.f - zg) * ng + zg * h_s[bb][u];
        }
    }
    __syncthreads();
    if (tid < 1200) {
        int bb = tid / 150, u = tid % 150;
        out[bb * 150 + u] = h_s[bb][u];
    }
}

// ---------------------------------------------------------------------------
// Workspace layout (floats): xc 160000 | si 320000 | sj 320000 |
// hcat 480000 | Gi 720000   => 2,000,000 floats = 8 MB total.
// ---------------------------------------------------------------------------
extern "C" void kernel_launch(void* const* d_in, const int* in_sizes, int n_in,
                              void* d_out, int out_size, void* d_ws, size_t ws_size,
                              hipStream_t stream)
{
    (void)in_sizes; (void)n_in; (void)out_size; (void)ws_size;
    const float* x       = (const float*)d_in[0];
    const float* conv_w  = (const float*)d_in[1];
    const float* conv_b  = (const float*)d_in[2];
    const float* f_lin_w = (const float*)d_in[3];
    const float* f_lin_b = (const float*)d_in[4];
    const float* f_a     = (const float*)d_in[5];
    const float* f_bias  = (const float*)d_in[6];
    const float* t_lin_w = (const float*)d_in[7];
    const float* t_lin_b = (const float*)d_in[8];
    const float* t_a     = (const float*)d_in[9];
    const float* t_bias  = (const float*)d_in[10];
    const float* w_ih    = (const float*)d_in[11];
    const float* w_hh    = (const float*)d_in[12];
    const float* b_ih    = (const float*)d_in[13];
    const float* b_hh    = (const float*)d_in[14];
    float* out = (float*)d_out;
    float* ws  = (float*)d_ws;

    float* xc   = ws;                 // (8,200,100)
    float* si   = ws + 160000;        // reused by both GATs
    float* sj   = ws + 480000;
    float* hcat = ws + 800000;        // (8,200,300)
    float* Gi   = ws + 1280000;       // (8,200,450)

    conv_relu_kernel<<<dim3((160000 + 255) / 256), dim3(256), 0, stream>>>(x, conv_w, conv_b, xc, hcat);

    // --- Feature GAT: v = xc^T (N=100 nodes, D=200). si includes f_lin_b.
    {
        int tiles = 7 * 25;                                  // ceil(100/16)*ceil(400/16)
        dim3 g((tiles + 7) / 8, 8);
        gemm_wmma_kernel<1, 0><<<g, 256, 0, stream>>>(xc, f_lin_w,           f_lin_b, si,
            100, 400, 200, 100, 400, 400, 20000, 0, 40000);
        gemm_wmma_kernel<1, 0><<<g, 256, 0, stream>>>(xc, f_lin_w + 200*400, nullptr, sj,
            100, 400, 200, 100, 400, 400, 20000, 0, 40000);
    }
    // out[b][i][d] -> hcat[b, d, 100+i]  (writes h_feat pre-transposed)
    gat_attn_kernel<<<dim3(100, 8), 256, 0, stream>>>(si, sj, f_a, f_bias, xc, hcat + 100,
        100, 400, 200, 40000, 20000, /*vJs*/1, /*vDs*/100, 60000, /*outIs*/1, /*outDs*/300);

    // --- Temporal GAT: v = xc (N=200, D=100).
    {
        int tiles = 13 * 13;                                 // ceil(200/16)^2
        dim3 g((tiles + 7) / 8, 8);
        gemm_wmma_kernel<0, 0><<<g, 256, 0, stream>>>(xc, t_lin_w,           t_lin_b, si,
            200, 200, 100, 100, 200, 200, 20000, 0, 40000);
        gemm_wmma_kernel<0, 0><<<g, 256, 0, stream>>>(xc, t_lin_w + 100*200, nullptr, sj,
            200, 200, 100, 100, 200, 200, 20000, 0, 40000);
    }
    // out[b][i][d] -> hcat[b, i, 200+d]
    gat_attn_kernel<<<dim3(200, 8), 256, 0, stream>>>(si, sj, t_a, t_bias, xc, hcat + 200,
        200, 200, 100, 40000, 20000, /*vJs*/100, /*vDs*/1, 60000, /*outIs*/300, /*outDs*/1);

    // --- GRU input gates for all timesteps: Gi = h_cat @ w_ih^T + b_ih
    {
        int tiles = 13 * 29;                                 // ceil(200/16)*ceil(450/16)
        dim3 g((tiles + 7) / 8, 8);
        gemm_wmma_kernel<0, 1><<<g, 256, 0, stream>>>(hcat, w_ih, b_ih, Gi,
            200, 450, 300, 300, 300, 450, 60000, 0, 90000);
    }

    // --- Sequential recurrence: one persistent workgroup, 29 waves.
    gru_wmma_kernel<<<1, 928, 0, stream>>>(Gi, w_hh, b_hh, out);
}